// RDFNSMultiHeadAttention_69913477644290
// MI455X (gfx1250) — compile-verified
//
#include <hip/hip_runtime.h>
#include <hip/hip_bf16.h>

// Problem constants (from reference)
#define BB 2
#define LL 1024
#define DD 1024
#define HH 16
#define HD 64
#define NEG_EXP (-65.5f)       // -(HD + ALPHA)
#define INV_SCALE (1.0f/64.0f) // 1/DIST_SCALE, BANDWIDTH=1
#define EPS12 1e-12f

typedef __attribute__((ext_vector_type(2))) float v2f;
typedef __attribute__((ext_vector_type(8))) float v8f;
typedef int v4i __attribute__((vector_size(16)));
typedef __attribute__((address_space(1))) void as1_void;
typedef __attribute__((address_space(3))) void as3_void;
typedef __attribute__((address_space(1))) v4i* v4i_gp;   // global b128 ptr
typedef __attribute__((address_space(3))) v4i* v4i_lp;   // LDS b128 ptr

__device__ __forceinline__ v8f wmma4(v2f a, v2f b, v8f c) {
    // V_WMMA_F32_16X16X4_F32 : D = A(16x4,f32) * B(4x16,f32) + C(16x16,f32)
    return __builtin_amdgcn_wmma_f32_16x16x4_f32(
        /*neg_a=*/false, a, /*neg_b=*/false, b,
        /*c_mod=*/(short)0, c, /*reuse_a=*/false, /*reuse_b=*/false);
}

// power-law kernel weight from squared distance
__device__ __forceinline__ float score_from_d2(float d2) {
    float g = __builtin_sqrtf(fmaxf(d2, EPS12)) * INV_SCALE;
    return exp2f(NEG_EXP * log2f(1.0f + g));
}

// ---- CDNA5 async global->LDS copy (ASYNCcnt path) with sync fallback ------
#if __has_builtin(__builtin_amdgcn_global_load_async_to_lds_b128)
#define HAVE_ASYNC_LDS 1
__device__ __forceinline__ void async_cp16(const float* g, float* l) {
    __builtin_amdgcn_global_load_async_to_lds_b128(
        (v4i_gp)(as1_void*)g, (v4i_lp)(as3_void*)l, /*offset=*/0, /*cpol=*/0);
}
#else
#define HAVE_ASYNC_LDS 0
__device__ __forceinline__ void async_cp16(const float* g, float* l) {
    *(float4*)l = *(const float4*)g;      // sync fallback: b128 load + ds_store
}
#endif

__device__ __forceinline__ void wait_async0() {
#if HAVE_ASYNC_LDS
#if __has_builtin(__builtin_amdgcn_s_wait_asynccnt)
    __builtin_amdgcn_s_wait_asynccnt(0);
#else
    asm volatile("s_wait_asynccnt 0x0" ::: "memory");
#endif
#endif
}

// ---------------------------------------------------------------------------
// Block-cooperative fp32 WMMA GEMM: C = A[MxK] @ B[KxN] + bias.
// 256 threads = 8 waves compute a 128x64 macro-tile (waves 4x2, 32x32 each).
// K-panels of 16 double-buffered in LDS via async global->LDS b128 copies.
// LDS row strides padded (17 / 68 floats) for conflict-free fragment reads.
// heads_layout=1 stores C[m,n] -> [B,H,L,HD]; else row-major [M,N].
// ---------------------------------------------------------------------------
#define GBM 128
#define GBN 64
#define GBK 16
#define AST 17
#define BST 68

__global__ __launch_bounds__(256) void gemm_wmma_f32(
    const float* __restrict__ A, const float* __restrict__ Bm,
    const float* __restrict__ bias, float* __restrict__ C,
    int M, int N, int K, int heads_layout)
{
    __shared__ float As[2][GBM * AST];   // 2 x 8704 B
    __shared__ float Bs[2][GBK * BST];   // 2 x 4352 B

    const int t    = threadIdx.x;
    const int wv   = t >> 5;
    const int lane = t & 31;
    const int half = lane >> 4;
    const int r    = lane & 15;
    const int wm   = wv >> 1;            // 0..3
    const int wn   = wv & 1;             // 0..1

    const int nb  = N / GBN;
    const int bm0 = (blockIdx.x / nb) * GBM;
    const int bn0 = (blockIdx.x % nb) * GBN;

    // cooperative loader indices
    const int ar = t >> 1, ac = (t & 1) * 8;    // A: 128 rows x 16 cols, 8 floats/thread
    const int br = t >> 4, bc = (t & 15) * 4;   // B: 16 rows x 64 cols, 4 floats/thread

    const float* Ag = A + (size_t)(bm0 + ar) * K + ac;
    const float* Bg = Bm + (size_t)br * N + bn0 + bc;

    auto load_panel = [&](int k0, int buf) {
        async_cp16(Ag + k0,     &As[buf][ar * AST + ac]);
        async_cp16(Ag + k0 + 4, &As[buf][ar * AST + ac + 4]);
        async_cp16(Bg + (size_t)k0 * N, &Bs[buf][br * BST + bc]);
    };

    v8f c00 = {}, c01 = {}, c10 = {}, c11 = {};
    const int nkb = K / GBK;
    load_panel(0, 0);

    for (int kb = 0; kb < nkb; ++kb) {
        const int buf = kb & 1;
        wait_async0();
        __syncthreads();                 // all waves' panel in LDS; prev reads done
        if (kb + 1 < nkb) load_panel((kb + 1) * GBK, buf ^ 1);

        const float* as0 = &As[buf][(wm * 32 + r) * AST];
        const float* as1 = as0 + 16 * AST;
        #pragma unroll
        for (int ks = 0; ks < 4; ++ks) {
            const int kk = ks * 4 + 2 * half;
            v2f a0 = *(const v2f*)(as0 + kk);
            v2f a1 = *(const v2f*)(as1 + kk);
            const float* bp = &Bs[buf][kk * BST + wn * 32 + r];
            v2f b0; b0.x = bp[0];  b0.y = bp[BST];
            v2f b1; b1.x = bp[16]; b1.y = bp[16 + BST];
            c00 = wmma4(a0, b0, c00);
            c01 = wmma4(a0, b1, c01);
            c10 = wmma4(a1, b0, c10);
            c11 = wmma4(a1, b1, c11);
        }
    }

    const int m0 = bm0 + wm * 32;
    const int n0 = bn0 + wn * 32;
    const float bv0 = bias[n0 + r];
    const float bv1 = bias[n0 + 16 + r];

    #pragma unroll
    for (int v = 0; v < 8; ++v) {
        const int rows[2] = { m0 + v + 8 * half, m0 + 16 + v + 8 * half };
        const float vals[2][2] = { { c00[v] + bv0, c01[v] + bv1 },
                                   { c10[v] + bv0, c11[v] + bv1 } };
        #pragma unroll
        for (int i = 0; i < 2; ++i) {
            const int m = rows[i];
            #pragma unroll
            for (int j = 0; j < 2; ++j) {
                const int n = n0 + j * 16 + r;
                size_t dst;
                if (heads_layout) {
                    const int b = m >> 10, l = m & (LL - 1);
                    const int h = n >> 6, hd = n & (HD - 1);
                    dst = (((size_t)(b * HH + h) * LL) + l) * HD + hd;
                } else {
                    dst = (size_t)m * N + n;
                }
                C[dst] = vals[i][j];
            }
        }
    }
}

// ---------------------------------------------------------------------------
// Per-row sum of squares over HD=64 (rows of [B*H*L, 64])
// ---------------------------------------------------------------------------
__global__ void rowsumsq_kernel(const float* __restrict__ X,
                                float* __restrict__ out, int nrows)
{
    const int row = blockIdx.x * blockDim.x + threadIdx.x;
    if (row >= nrows) return;
    const float4* p = (const float4*)(X + (size_t)row * HD);
    float s = 0.f;
    #pragma unroll
    for (int i = 0; i < HD / 4; ++i) {
        float4 q = p[i];
        s += q.x * q.x + q.y * q.y + q.z * q.z + q.w * q.w;
    }
    out[row] = s;
}

// ---------------------------------------------------------------------------
// Pass A: column sums of the score matrix; outputs ncs[j] = N_C[j]^(-0.5).
// One wave per (bh, 16-column strip); loops over all 64 row tiles.
// ---------------------------------------------------------------------------
__global__ __launch_bounds__(256) void colsum_kernel(
    const float* __restrict__ Q, const float* __restrict__ Kc,
    const float* __restrict__ q2, const float* __restrict__ k2,
    float* __restrict__ ncs)
{
    const int wave = blockIdx.x * 8 + (threadIdx.x >> 5);
    const int lane = threadIdx.x & 31;
    const int half = lane >> 4;
    const int r    = lane & 15;
    const int jtiles = LL / 16;
    const int bh = wave / jtiles;
    const int j0 = (wave - bh * jtiles) * 16;

    const float* Qb = Q  + (size_t)bh * LL * HD;
    const float* Kb = Kc + (size_t)bh * LL * HD;

    // B fragments of K^T for this column strip: B[c][j] = K[j][c]
    v2f bf[16];
    const float* krow = Kb + (size_t)(j0 + r) * HD;
    #pragma unroll
    for (int ks = 0; ks < 16; ++ks)
        bf[ks] = *(const v2f*)(krow + ks * 4 + 2 * half);

    const float k2v = k2[bh * LL + j0 + r];
    float cs = 0.f;

    for (int i0 = 0; i0 < LL; i0 += 16) {
        v8f qk = {};
        const float* qrow = Qb + (size_t)(i0 + r) * HD;
        #pragma unroll
        for (int ks = 0; ks < 16; ++ks) {
            v2f af = *(const v2f*)(qrow + ks * 4 + 2 * half);
            qk = wmma4(af, bf[ks], qk);
        }
        const float* q2p = q2 + bh * LL + i0 + 8 * half;
        #pragma unroll
        for (int v = 0; v < 8; ++v) {
            float d2 = q2p[v] + k2v - 2.0f * qk[v];
            cs += score_from_d2(d2);
        }
    }
    cs += __shfl_xor(cs, 16);          // combine halves -> full column sum
    if (half == 0)
        ncs[bh * LL + j0 + r] = __frsqrt_rn(cs);   // N_C^(-a), a = 0.5
}

// ---------------------------------------------------------------------------
// Pass B: fused attention. One wave per (bh, 16-row strip).
// out_i = (sum_j w_ij * V_j) / max(sum_j w_ij, eps),  w_ij = S_ij * ncs[j].
// W tile bounced through LDS (C-layout -> A-layout) for the W@V WMMA.
// ---------------------------------------------------------------------------
__global__ __launch_bounds__(256) void attn_kernel(
    const float* __restrict__ Q, const float* __restrict__ Kc,
    const float* __restrict__ V, const float* __restrict__ q2,
    const float* __restrict__ k2, const float* __restrict__ ncs,
    float* __restrict__ outw)
{
    __shared__ float lw[8 * 256];      // 1KB (16x16 f32) per wave
    const int wv   = threadIdx.x >> 5;
    const int wave = blockIdx.x * 8 + wv;
    const int lane = threadIdx.x & 31;
    const int half = lane >> 4;
    const int r    = lane & 15;
    const int itiles = LL / 16;
    const int bh = wave / itiles;
    const int i0 = (wave - bh * itiles) * 16;
    const int b  = bh >> 4;
    const int h  = bh & 15;

    const float* Qb = Q  + (size_t)bh * LL * HD;
    const float* Kb = Kc + (size_t)bh * LL * HD;
    const float* Vb = V  + (size_t)bh * LL * HD;
    float* lws = lw + wv * 256;

    // Preload Q A-fragments for all 16 K-steps (reused over all 64 j tiles)
    v2f qa[16];
    const float* qrow = Qb + (size_t)(i0 + r) * HD;
    #pragma unroll
    for (int ks = 0; ks < 16; ++ks)
        qa[ks] = *(const v2f*)(qrow + ks * 4 + 2 * half);

    float q2v[8];
    #pragma unroll
    for (int v = 0; v < 8; ++v)
        q2v[v] = q2[bh * LL + i0 + v + 8 * half];

    v8f acc0 = {}, acc1 = {}, acc2 = {}, acc3 = {};
    float rs[8] = {0.f, 0.f, 0.f, 0.f, 0.f, 0.f, 0.f, 0.f};

    for (int j0 = 0; j0 < LL; j0 += 16) {
        // ---- distances for this 16x16 tile
        v8f qk = {};
        const float* krow = Kb + (size_t)(j0 + r) * HD;
        #pragma unroll
        for (int ks = 0; ks < 16; ++ks) {
            v2f bf = *(const v2f*)(krow + ks * 4 + 2 * half);
            qk = wmma4(qa[ks], bf, qk);
        }
        const float k2v = k2[bh * LL + j0 + r];
        const float nv  = ncs[bh * LL + j0 + r];
        #pragma unroll
        for (int v = 0; v < 8; ++v) {
            float d2 = q2v[v] + k2v - 2.0f * qk[v];
            float w  = score_from_d2(d2) * nv;
            rs[v] += w;
            lws[(v + 8 * half) * 16 + r] = w;   // C-layout -> LDS
        }
        // ---- W @ V  (K=16 -> 4 WMMA K-steps, N=64 -> 4 accumulators)
        #pragma unroll
        for (int ks = 0; ks < 4; ++ks) {
            v2f aw = *(const v2f*)(lws + r * 16 + ks * 4 + 2 * half);
            const int jj = j0 + ks * 4 + 2 * half;
            const float* vp = Vb + (size_t)jj * HD + r;
            v2f b0; b0.x = vp[0];  b0.y = vp[HD];
            v2f b1; b1.x = vp[16]; b1.y = vp[16 + HD];
            v2f b2; b2.x = vp[32]; b2.y = vp[32 + HD];
            v2f b3; b3.x = vp[48]; b3.y = vp[48 + HD];
            acc0 = wmma4(aw, b0, acc0);
            acc1 = wmma4(aw, b1, acc1);
            acc2 = wmma4(aw, b2, acc2);
            acc3 = wmma4(aw, b3, acc3);
        }
    }

    // Row-sum reduction within each 16-lane half, then normalize + store
    #pragma unroll
    for (int v = 0; v < 8; ++v) {
        float s = rs[v];
        s += __shfl_xor(s, 1, 16);
        s += __shfl_xor(s, 2, 16);
        s += __shfl_xor(s, 4, 16);
        s += __shfl_xor(s, 8, 16);
        rs[v] = 1.0f / fmaxf(s, EPS12);
    }
    #pragma unroll
    for (int v = 0; v < 8; ++v) {
        const int l = i0 + v + 8 * half;
        float* orow = outw + ((size_t)(b * LL + l)) * DD + h * HD;
        orow[r]      = acc0[v] * rs[v];
        orow[16 + r] = acc1[v] * rs[v];
        orow[32 + r] = acc2[v] * rs[v];
        orow[48 + r] = acc3[v] * rs[v];
    }
}

// ---------------------------------------------------------------------------
extern "C" void kernel_launch(void* const* d_in, const int* in_sizes, int n_in,
                              void* d_out, int out_size, void* d_ws, size_t ws_size,
                              hipStream_t stream) {
    const float* x  = (const float*)d_in[0];
    const float* WQ = (const float*)d_in[1];
    const float* bQ = (const float*)d_in[2];
    const float* WK = (const float*)d_in[3];
    const float* bK = (const float*)d_in[4];
    const float* WV = (const float*)d_in[5];
    const float* bV = (const float*)d_in[6];
    const float* WO = (const float*)d_in[7];
    const float* bO = (const float*)d_in[8];
    float* out = (float*)d_out;

    const size_t NQ = (size_t)BB * HH * LL * HD;   // 2,097,152
    const size_t NR = (size_t)BB * HH * LL;        // 32,768
    float* ws  = (float*)d_ws;
    float* q   = ws;
    float* k   = q  + NQ;
    float* v   = k  + NQ;
    float* aw  = v  + NQ;        // merged-heads attention output [B,L,D]
    float* q2  = aw + NQ;
    float* k2  = q2 + NR;
    float* ncs = k2 + NR;

    const int M = BB * LL;       // 2048
    // 128x64 macro-tiles: (2048/128)*(1024/64) = 256 blocks of 8 waves
    const int gblocks = (M / GBM) * (DD / GBN);
    gemm_wmma_f32<<<gblocks, 256, 0, stream>>>(x, WQ, bQ, q, M, DD, DD, 1);
    gemm_wmma_f32<<<gblocks, 256, 0, stream>>>(x, WK, bK, k, M, DD, DD, 1);
    gemm_wmma_f32<<<gblocks, 256, 0, stream>>>(x, WV, bV, v, M, DD, DD, 1);

    rowsumsq_kernel<<<(int)(NR / 256), 256, 0, stream>>>(q, q2, (int)NR);
    rowsumsq_kernel<<<(int)(NR / 256), 256, 0, stream>>>(k, k2, (int)NR);

    // 2*16 heads * 64 strips = 2048 waves -> 256 blocks
    colsum_kernel<<<256, 256, 0, stream>>>(q, k, q2, k2, ncs);
    attn_kernel<<<256, 256, 0, stream>>>(q, k, v, q2, k2, ncs, aw);

    gemm_wmma_f32<<<gblocks, 256, 0, stream>>>(aw, WO, bO, out, M, DD, DD, 0);
}